// SSIMLoss_13993003450993
// MI455X (gfx1250) — compile-verified
//
#include <hip/hip_runtime.h>
#include <math.h>

typedef __attribute__((ext_vector_type(2))) float v2f;
typedef __attribute__((ext_vector_type(8))) float v8f;

#define IMG_H 512
#define IMG_W 512
#define NBATCH 32
#define TILE 16
#define HALO 5
#define KW 11
#define PADDIM 28      // 16 + 2*5 = 26, padded to 28 (multiple of 4 for K-chunking)
#define NCHUNK 7       // 28 / 4
#define NPIX (NBATCH * IMG_H * IMG_W)

// sum_{i=0..10} exp(-(i-5)^2 / (2*1.5^2)) = 3.75923278...
#define GAUSS_INV_NORM 0.26601478f
#define GAUSS_NEG_INV_2S2 (-1.0f / 4.5f)

// ---------------- workspace init ----------------
__global__ void init_ws_kernel(unsigned int* mm, float* sum) {
    mm[0] = 0x7F800000u;  // +inf bits  (running min, values are >= 0)
    mm[1] = 0u;           // 0.0f bits  (running max, values are >= 0)
    *sum = 0.f;
}

// ---------------- target min/max (for dynamic data_range) ----------------
__global__ void minmax_kernel(const float* __restrict__ t, unsigned int* mm, int n) {
    float lmin = INFINITY, lmax = -INFINITY;
    for (int i = blockIdx.x * blockDim.x + threadIdx.x; i < n;
         i += gridDim.x * blockDim.x) {
        float v = t[i];
        lmin = fminf(lmin, v);
        lmax = fmaxf(lmax, v);
    }
    for (int off = 16; off > 0; off >>= 1) {
        lmin = fminf(lmin, __shfl_down(lmin, off, 32));
        lmax = fmaxf(lmax, __shfl_down(lmax, off, 32));
    }
    if ((threadIdx.x & 31) == 0) {
        // positive floats order-preserve as signed ints; -inf/+inf inits also safe
        atomicMin((int*)&mm[0], __float_as_int(lmin));
        atomicMax((int*)&mm[1], __float_as_int(lmax));
    }
}

// ---------------- fused SSIM tile kernel (WMMA f32 16x16x4) ----------------
__global__ __launch_bounds__(32)
void ssim_tile_kernel(const float* __restrict__ pred,
                      const float* __restrict__ tgt,
                      const unsigned int* __restrict__ mm,
                      float* __restrict__ sum_out) {
    __shared__ float sP[PADDIM * PADDIM];
    __shared__ float sT[PADDIM * PADDIM];
    __shared__ float sH[5][PADDIM][TILE];  // horizontally blurred quantities

    const int lane = threadIdx.x;
    const int tc = blockIdx.x * TILE;
    const int tr = blockIdx.y * TILE;
    const int b  = blockIdx.z;
    const float* pbase = pred + (size_t)b * IMG_H * IMG_W;
    const float* tbase = tgt  + (size_t)b * IMG_H * IMG_W;

    // load 28x28 halo tiles (zero padding == conv SAME)
    for (int idx = lane; idx < PADDIM * PADDIM; idx += 32) {
        int rr = idx / PADDIM, cc = idx % PADDIM;
        int gr = tr - HALO + rr, gc = tc - HALO + cc;
        bool ok = (gr >= 0) && (gr < IMG_H) && (gc >= 0) && (gc < IMG_W);
        size_t off = (size_t)gr * IMG_W + gc;
        sP[idx] = ok ? pbase[off] : 0.f;
        sT[idx] = ok ? tbase[off] : 0.f;
    }
    __syncthreads();

    const int n   = lane & 15;              // output col (and M for vertical A)
    const int kk0 = (lane < 16) ? 0 : 2;    // K phase within a 4-chunk
    const int half8 = (lane < 16) ? 0 : 8;  // C/D row offset for this half-wave

    // Banded Gaussian fragments: G[k][n] = g[k-n]; identical banded structure is
    // the horizontal B matrix and the vertical A matrix. Computed arithmetically
    // (one v_exp per value) instead of a dynamic-index select ladder.
    v2f gfrag[NCHUNK];
#pragma unroll
    for (int t = 0; t < NCHUNK; ++t) {
#pragma unroll
        for (int u = 0; u < 2; ++u) {
            int d = 4 * t + kk0 + u - n;      // tap index into g[0..10]
            float x = (float)(d - KW / 2);    // distance from window center
            float w = __expf(x * x * GAUSS_NEG_INV_2S2) * GAUSS_INV_NORM;
            gfrag[t][u] = (d >= 0 && d < KW) ? w : 0.f;
        }
    }

    // ---- horizontal pass: H[row][n] for rows 0..27 (two overlapping 16-row groups)
#pragma unroll
    for (int rg = 0; rg < 2; ++rg) {
        const int rb = rg * 12;  // rows rb .. rb+15
        v8f a0 = {}, a1 = {}, a2 = {}, a3 = {}, a4 = {};
#pragma unroll
        for (int t = 0; t < NCHUNK; ++t) {
            const int row = rb + n;
            const int c0 = 4 * t + kk0;
            float p0 = sP[row * PADDIM + c0], p1 = sP[row * PADDIM + c0 + 1];
            float q0 = sT[row * PADDIM + c0], q1 = sT[row * PADDIM + c0 + 1];
            v2f Ap  = {p0, p1};
            v2f At  = {q0, q1};
            v2f App = {p0 * p0, p1 * p1};
            v2f Att = {q0 * q0, q1 * q1};
            v2f Apt = {p0 * q0, p1 * q1};
            v2f Bw = gfrag[t];
            a0 = __builtin_amdgcn_wmma_f32_16x16x4_f32(false, Ap,  false, Bw, (short)0, a0, false, false);
            a1 = __builtin_amdgcn_wmma_f32_16x16x4_f32(false, At,  false, Bw, (short)0, a1, false, false);
            a2 = __builtin_amdgcn_wmma_f32_16x16x4_f32(false, App, false, Bw, (short)0, a2, false, false);
            a3 = __builtin_amdgcn_wmma_f32_16x16x4_f32(false, Att, false, Bw, (short)0, a3, false, false);
            a4 = __builtin_amdgcn_wmma_f32_16x16x4_f32(false, Apt, false, Bw, (short)0, a4, false, false);
        }
#pragma unroll
        for (int j = 0; j < 8; ++j) {
            const int m = rb + j + half8;  // rg0: rows 0..15, rg1: rows 12..27
            sH[0][m][n] = a0[j];
            sH[1][m][n] = a1[j];
            sH[2][m][n] = a2[j];
            sH[3][m][n] = a3[j];
            sH[4][m][n] = a4[j];
        }
    }
    __syncthreads();

    // ---- vertical pass: V = bandedGauss(16x28) * H(28x16)
    v8f m0 = {}, m1 = {}, m2 = {}, m3 = {}, m4 = {};
#pragma unroll
    for (int t = 0; t < NCHUNK; ++t) {
        const int k0 = 4 * t + kk0;
        v2f Aw = gfrag[t];
        v2f B0 = {sH[0][k0][n], sH[0][k0 + 1][n]};
        v2f B1 = {sH[1][k0][n], sH[1][k0 + 1][n]};
        v2f B2 = {sH[2][k0][n], sH[2][k0 + 1][n]};
        v2f B3 = {sH[3][k0][n], sH[3][k0 + 1][n]};
        v2f B4 = {sH[4][k0][n], sH[4][k0 + 1][n]};
        m0 = __builtin_amdgcn_wmma_f32_16x16x4_f32(false, Aw, false, B0, (short)0, m0, false, false);
        m1 = __builtin_amdgcn_wmma_f32_16x16x4_f32(false, Aw, false, B1, (short)0, m1, false, false);
        m2 = __builtin_amdgcn_wmma_f32_16x16x4_f32(false, Aw, false, B2, (short)0, m2, false, false);
        m3 = __builtin_amdgcn_wmma_f32_16x16x4_f32(false, Aw, false, B3, (short)0, m3, false, false);
        m4 = __builtin_amdgcn_wmma_f32_16x16x4_f32(false, Aw, false, B4, (short)0, m4, false, false);
    }

    // ---- per-pixel SSIM and reduction
    const float dr = __uint_as_float(mm[1]) - __uint_as_float(mm[0]);
    float C1 = 0.01f * dr; C1 *= C1;
    float C2 = 0.03f * dr; C2 *= C2;

    float lsum = 0.f;
#pragma unroll
    for (int j = 0; j < 8; ++j) {
        float mux = m0[j], muy = m1[j];
        float exx = m2[j], eyy = m3[j], exy = m4[j];
        float mux2 = mux * mux, muy2 = muy * muy, muxy = mux * muy;
        float sx = exx - mux2, sy = eyy - muy2, sxy = exy - muxy;
        float num = (2.f * muxy + C1) * (2.f * sxy + C2);
        float den = (mux2 + muy2 + C1) * (sx + sy + C2);
        lsum += num / den;
    }
    for (int off = 16; off > 0; off >>= 1) lsum += __shfl_down(lsum, off, 32);
    if (lane == 0) atomicAdd(sum_out, lsum);
}

// ---------------- finalize ----------------
__global__ void finalize_kernel(const float* sum, float* out) {
    out[0] = 1.f - sum[0] * (1.f / (float)NPIX);
}

extern "C" void kernel_launch(void* const* d_in, const int* in_sizes, int n_in,
                              void* d_out, int out_size, void* d_ws, size_t ws_size,
                              hipStream_t stream) {
    const float* pred = (const float*)d_in[0];
    const float* tgt  = (const float*)d_in[1];
    float* out = (float*)d_out;
    unsigned int* mm = (unsigned int*)d_ws;      // [0]=min bits, [1]=max bits
    float* sum = (float*)d_ws + 2;               // [2]=ssim_map sum

    init_ws_kernel<<<1, 1, 0, stream>>>(mm, sum);
    minmax_kernel<<<512, 256, 0, stream>>>(tgt, mm, NPIX);

    dim3 grid(IMG_W / TILE, IMG_H / TILE, NBATCH);  // 32 x 32 x 32 tiles
    ssim_tile_kernel<<<grid, 32, 0, stream>>>(pred, tgt, mm, sum);

    finalize_kernel<<<1, 1, 0, stream>>>(sum, out);
}